// IntegratedBoundingBoxModel_74457553044299
// MI455X (gfx1250) — compile-verified
//
#include <hip/hip_runtime.h>
#include <hip/hip_bf16.h>
#include <stdint.h>

// ---- problem constants -----------------------------------------------------
#define B_    16
#define R_    2048
#define C_    81
#define Q_    65
#define D_    1024
#define K_    1024
#define DET_  100
#define M_    (R_ * (C_ - 1))          // 163840 candidates / image
#define IMG_  640.0f
#define OFFS_ 641.0f                   // IMG + 1 per-class NMS offset
#define SCORE_TH 0.05f
#define NMS_TH   0.5f
#define MIN_SZ   0.01f
#define BBOX_CLIP_ 4.135166556742356f  // log(1000/16)

#if __has_builtin(__builtin_amdgcn_global_load_async_to_lds_b128)
#define HAVE_ASYNC_B128 1
typedef __attribute__((ext_vector_type(4))) int v4i_;
typedef __attribute__((address_space(1))) v4i_ gv4i_;   // global (AS1)
typedef __attribute__((address_space(3))) v4i_ lv4i_;   // LDS (AS3)
#endif

static __device__ __forceinline__ void wait_async_zero() {
#if __has_builtin(__builtin_amdgcn_s_wait_asynccnt)
  __builtin_amdgcn_s_wait_asynccnt(0);
#else
  asm volatile("s_wait_asynccnt 0" ::: "memory");
#endif
}

// torchvision BoxCoder.decode + clip to [0, IMG]
static __device__ __forceinline__ void decode_clip(const float* __restrict__ rg,
                                                   float p0, float p1, float p2, float p3,
                                                   float bx[4]) {
  float pw  = p2 - p0, ph = p3 - p1;
  float pcx = p0 + 0.5f * pw, pcy = p1 + 0.5f * ph;
  float dx = rg[0] * 0.1f, dy = rg[1] * 0.1f;
  float dw = fminf(rg[2] * 0.2f, BBOX_CLIP_);
  float dh = fminf(rg[3] * 0.2f, BBOX_CLIP_);
  float cx = dx * pw + pcx, cy = dy * ph + pcy;
  float w = expf(dw) * pw, h = expf(dh) * ph;
  bx[0] = fminf(fmaxf(cx - 0.5f * w, 0.0f), IMG_);
  bx[1] = fminf(fmaxf(cy - 0.5f * h, 0.0f), IMG_);
  bx[2] = fminf(fmaxf(cx + 0.5f * w, 0.0f), IMG_);
  bx[3] = fminf(fmaxf(cy + 0.5f * h, 0.0f), IMG_);
}

// ---- kernel 1: softmax + decode validity -> sortable keys; quantity argmax -
// one wave32 per proposal row
__global__ __launch_bounds__(256) void k1_score(
    const float* __restrict__ logits, const float* __restrict__ qlog,
    const float* __restrict__ reg, const float* __restrict__ props,
    unsigned* __restrict__ keys, int* __restrict__ quant) {
  int wid  = (int)((blockIdx.x * blockDim.x + threadIdx.x) >> 5);
  int lane = threadIdx.x & 31;
  if (wid >= B_ * R_) return;
  int b = wid / R_, r = wid % R_;
  const float* lg = logits + (size_t)wid * C_;
  const float* rg = reg    + (size_t)wid * (C_ * 4);
  const float* pp = props  + (size_t)wid * 4;
  const float* ql = qlog   + (size_t)wid * Q_;
  __builtin_prefetch(rg, 0, 1);
  __builtin_prefetch(rg + 64, 0, 1);
  __builtin_prefetch(rg + 128, 0, 1);

  // softmax: lanes stride classes
  float lv[3];
  float mx = -3.4e38f;
  {
    int i = 0;
    for (int c = lane; c < C_; c += 32) { lv[i] = lg[c]; mx = fmaxf(mx, lv[i]); ++i; }
  }
  for (int o = 16; o; o >>= 1) mx = fmaxf(mx, __shfl_xor(mx, o, 32));
  float sum = 0.0f;
  {
    int i = 0;
    for (int c = lane; c < C_; c += 32) { sum += expf(lv[i] - mx); ++i; }
  }
  for (int o = 16; o; o >>= 1) sum += __shfl_xor(sum, o, 32);
  float inv = 1.0f / sum;

  float p0 = pp[0], p1 = pp[1], p2 = pp[2], p3 = pp[3];
  {
    int i = 0;
    for (int c = lane; c < C_; c += 32) {
      if (c >= 1) {
        float s = expf(lv[i] - mx) * inv;
        float bx[4];
        decode_clip(rg + 4 * c, p0, p1, p2, p3, bx);
        float bw = bx[2] - bx[0], bh = bx[3] - bx[1];
        unsigned key = 0u;
        if (s > SCORE_TH && bw >= MIN_SZ && bh >= MIN_SZ) key = __float_as_uint(s);
        keys[(size_t)b * M_ + (size_t)r * (C_ - 1) + (c - 1)] = key;
      }
      ++i;
    }
  }

  // quantity argmax (first-occurrence tie-break)
  float qv = -3.4e38f; int qi = 0x7fffffff;
  for (int q = lane; q < Q_; q += 32) {
    float v = ql[q];
    if (v > qv || (v == qv && q < qi)) { qv = v; qi = q; }
  }
  for (int o = 16; o; o >>= 1) {
    float v2 = __shfl_xor(qv, o, 32);
    int   i2 = __shfl_xor(qi, o, 32);
    if (v2 > qv || (v2 == qv && i2 < qi)) { qv = v2; qi = i2; }
  }
  if (lane == 0) quant[wid] = qi;
}

// ---- kernel 2: per-image radix-select top-K + bitonic sort + re-decode -----
__global__ __launch_bounds__(1024) void k2_select(
    const unsigned* __restrict__ keys, const float* __restrict__ reg,
    const float* __restrict__ props,
    float* __restrict__ candBox, float* __restrict__ obox,
    float* __restrict__ score, int* __restrict__ label,
    int* __restrict__ rowIdx, int* __restrict__ cvFlag) {
  int b = blockIdx.x, t = threadIdx.x;
  const unsigned* kb = keys + (size_t)b * M_;
  __shared__ unsigned hist[256];
  __shared__ unsigned sPick[2];
  __shared__ unsigned cnts[2];
  __shared__ unsigned long long sk[K_];

  unsigned prefixVal = 0, prefixMask = 0;
  int need = K_;
  for (int pass = 3; pass >= 0; --pass) {
    int shift = pass * 8;
    if (t < 256) hist[t] = 0u;
    __syncthreads();
    for (int j = t; j < M_; j += 1024) {
      __builtin_prefetch(kb + j + 8192, 0, 1);
      unsigned k = kb[j];
      if ((k & prefixMask) == prefixVal) atomicAdd(&hist[(k >> shift) & 0xFF], 1u);
    }
    __syncthreads();
    if (t == 0) {
      unsigned cum = 0; int d = 0;
      for (int dd = 255; dd >= 0; --dd) {
        unsigned h = hist[dd];
        if (cum + h >= (unsigned)need) { d = dd; break; }
        cum += h;
      }
      sPick[0] = (unsigned)d; sPick[1] = cum;
    }
    __syncthreads();
    unsigned d = sPick[0], cumG = sPick[1];
    need -= (int)cumG;
    prefixVal  |= d << shift;
    prefixMask |= 0xFFu << shift;
    __syncthreads();
  }
  unsigned T = prefixVal;
  int G = K_ - need;  // count of keys strictly greater than T (< K)

  if (t < 2) cnts[t] = 0u;
  __syncthreads();
  for (int j = t; j < M_; j += 1024) {
    unsigned k = kb[j];
    if (k > T) {
      unsigned p = atomicAdd(&cnts[0], 1u);
      sk[p] = ((unsigned long long)k << 32) | (unsigned)(~j);
    } else if (k == T) {
      unsigned e = atomicAdd(&cnts[1], 1u);
      if ((int)e < need)
        sk[G + (int)e] = ((unsigned long long)k << 32) | (unsigned)(~j);
    }
  }
  __syncthreads();

  // bitonic sort, descending (key desc, then index asc via ~idx)
  for (int sz = 2; sz <= K_; sz <<= 1) {
    for (int st = sz >> 1; st > 0; st >>= 1) {
      int l = t ^ st;
      if (l > t) {
        unsigned long long a = sk[t], c = sk[l];
        bool descBlock = ((t & sz) == 0);
        bool sw = descBlock ? (a < c) : (a > c);
        if (sw) { sk[t] = c; sk[l] = a; }
      }
      __syncthreads();
    }
  }

  unsigned long long v = sk[t];
  unsigned k = (unsigned)(v >> 32);
  int idx = (int)(~(unsigned)(v & 0xFFFFFFFFu));
  int r = idx / (C_ - 1);
  int c = idx - r * (C_ - 1) + 1;
  const float* rg = reg   + ((size_t)b * R_ + r) * (C_ * 4) + 4 * c;
  const float* pp = props + ((size_t)b * R_ + r) * 4;
  float bx[4];
  decode_clip(rg, pp[0], pp[1], pp[2], pp[3], bx);
  size_t o = (size_t)b * K_ + t;
  candBox[o * 4 + 0] = bx[0]; candBox[o * 4 + 1] = bx[1];
  candBox[o * 4 + 2] = bx[2]; candBox[o * 4 + 3] = bx[3];
  float off = (float)c * OFFS_;
  obox[o * 4 + 0] = bx[0] + off; obox[o * 4 + 1] = bx[1] + off;
  obox[o * 4 + 2] = bx[2] + off; obox[o * 4 + 3] = bx[3] + off;
  score[o]  = (k > 0u) ? __uint_as_float(k) : -1.0f;
  label[o]  = c;
  rowIdx[o] = r;
  cvFlag[o] = (k > 0u) ? 1 : 0;
}

// ---- kernel 3: per-image LDS-resident sequential NMS + det compaction ------
__global__ __launch_bounds__(1024) void k3_nms(
    const float* __restrict__ obox, const float* __restrict__ candBox,
    const float* __restrict__ score, const int* __restrict__ label,
    const int* __restrict__ rowIdx, const int* __restrict__ cvFlag,
    const int* __restrict__ quant,
    float* __restrict__ o_db, float* __restrict__ o_ds,
    int* __restrict__ o_dl, int* __restrict__ o_dq,
    int* __restrict__ o_dv, int* __restrict__ detRow) {
  int b = blockIdx.x, t = threadIdx.x;
  __shared__ __align__(16) float sB[K_][4];
  __shared__ float sArea[K_];
  __shared__ unsigned keepBits[K_ / 32];
  __shared__ int sc[K_];
  __shared__ int kcS;
  size_t o = (size_t)b * K_ + t;

  // stage all 1024 offset boxes into LDS (async data path on gfx1250)
#ifdef HAVE_ASYNC_B128
  __builtin_amdgcn_global_load_async_to_lds_b128(
      (gv4i_*)(obox + o * 4),
      (lv4i_*)&sB[t][0],
      0, 0);
  wait_async_zero();
#else
  {
    float4 bv = *(const float4*)(obox + o * 4);
    *(float4*)&sB[t][0] = bv;
  }
#endif

  int cv = cvFlag[o];
  if (t < K_ / 32) keepBits[t] = 0u;
  __syncthreads();
  sArea[t] = (sB[t][2] - sB[t][0]) * (sB[t][3] - sB[t][1]);
  if (cv) atomicOr(&keepBits[t >> 5], 1u << (t & 31));
  __syncthreads();

  float x1 = sB[t][0], y1 = sB[t][1], x2 = sB[t][2], y2 = sB[t][3];
  float ar = sArea[t];
  for (int i = 0; i < K_; ++i) {
    unsigned w = keepBits[i >> 5];
    if ((w >> (i & 31)) & 1u) {       // keep[i] & cv[i] (bits seeded from cv)
      if (t > i) {
        float ix1 = fmaxf(x1, sB[i][0]);
        float iy1 = fmaxf(y1, sB[i][1]);
        float ix2 = fminf(x2, sB[i][2]);
        float iy2 = fminf(y2, sB[i][3]);
        float iw = fmaxf(ix2 - ix1, 0.0f), ih = fmaxf(iy2 - iy1, 0.0f);
        float inter = iw * ih;
        float iou = inter / (ar + sArea[i] - inter + 1e-9f);
        if (iou > NMS_TH) atomicAnd(&keepBits[t >> 5], ~(1u << (t & 31)));
      }
    }
    __syncthreads();
  }

  int kept = (int)((keepBits[t >> 5] >> (t & 31)) & 1u);
  sc[t] = kept;
  __syncthreads();
  for (int off = 1; off < K_; off <<= 1) {
    int v = (t >= off) ? sc[t - off] : 0;
    __syncthreads();
    sc[t] += v;
    __syncthreads();
  }
  if (t == K_ - 1) kcS = sc[t];
  __syncthreads();
  int pos = sc[t] - kept;               // stable (score-sorted) output slot
  int kc = kcS; if (kc > DET_) kc = DET_;

  if (kept && pos < DET_) {
    size_t oo = (size_t)b * DET_ + pos;
    const float* cb = candBox + o * 4;
    o_db[oo * 4 + 0] = cb[0]; o_db[oo * 4 + 1] = cb[1];
    o_db[oo * 4 + 2] = cb[2]; o_db[oo * 4 + 3] = cb[3];
    o_ds[oo] = score[o];
    o_dl[oo] = label[o];
    int r = rowIdx[o];
    o_dq[oo] = quant[b * R_ + r];
    detRow[oo] = r;
  }
  if (t < DET_) {
    size_t oo = (size_t)b * DET_ + t;
    o_dv[oo] = (t < kc) ? 1 : 0;
    if (t >= kc) {
      o_db[oo * 4 + 0] = 0.0f; o_db[oo * 4 + 1] = 0.0f;
      o_db[oo * 4 + 2] = 0.0f; o_db[oo * 4 + 3] = 0.0f;
      o_ds[oo] = 0.0f; o_dl[oo] = 0; o_dq[oo] = 0;
      detRow[oo] = -1;
    }
  }
}

// ---- kernel 4: feature gather for surviving detections ---------------------
__global__ __launch_bounds__(256) void k4_feat(
    const float* __restrict__ feats, const int* __restrict__ detRow,
    float* __restrict__ o_df) {
  int slot = blockIdx.x;                 // b*DET + j
  int b = slot / DET_;
  int r = detRow[slot];
  int t = threadIdx.x;
  float4* dst = (float4*)(o_df + (size_t)slot * D_);
  if (r < 0) { dst[t] = make_float4(0.f, 0.f, 0.f, 0.f); return; }
  const float4* src = (const float4*)(feats + ((size_t)b * R_ + r) * D_);
  __builtin_prefetch(src + t, 0, 1);
  dst[t] = src[t];
}

// ---- host-side launch ------------------------------------------------------
extern "C" void kernel_launch(void* const* d_in, const int* in_sizes, int n_in,
                              void* d_out, int out_size, void* d_ws, size_t ws_size,
                              hipStream_t stream) {
  (void)in_sizes; (void)n_in; (void)out_size; (void)ws_size;
  const float* class_logits = (const float*)d_in[0];
  const float* qlog         = (const float*)d_in[1];
  const float* feats        = (const float*)d_in[2];
  const float* reg          = (const float*)d_in[3];
  const float* props        = (const float*)d_in[4];

  char* ws = (char*)d_ws;
  size_t off = 0;
  auto alloc = [&](size_t bytes) -> void* {
    void* p = ws + off;
    off += (bytes + 255) & ~(size_t)255;
    return p;
  };
  unsigned* keys   = (unsigned*)alloc((size_t)B_ * M_ * 4);
  int*      quant  = (int*)alloc((size_t)B_ * R_ * 4);
  float*    candBox= (float*)alloc((size_t)B_ * K_ * 16);
  float*    obox   = (float*)alloc((size_t)B_ * K_ * 16);
  float*    score  = (float*)alloc((size_t)B_ * K_ * 4);
  int*      label  = (int*)alloc((size_t)B_ * K_ * 4);
  int*      rowIdx = (int*)alloc((size_t)B_ * K_ * 4);
  int*      cvFlag = (int*)alloc((size_t)B_ * K_ * 4);
  int*      detRow = (int*)alloc((size_t)B_ * DET_ * 4);

  float* out  = (float*)d_out;
  float* o_db = out;                                   // [B,DET,4]
  float* o_ds = out + (size_t)B_ * DET_ * 4;           // [B,DET]
  int*   o_dl = (int*)(out + (size_t)B_ * DET_ * 5);   // [B,DET] i32
  int*   o_dq = (int*)(out + (size_t)B_ * DET_ * 6);   // [B,DET] i32
  float* o_df = out + (size_t)B_ * DET_ * 7;           // [B,DET,D]
  int*   o_dv = (int*)(o_df + (size_t)B_ * DET_ * D_); // [B,DET] valid

  k1_score<<<(B_ * R_) / 8, 256, 0, stream>>>(class_logits, qlog, reg, props, keys, quant);
  k2_select<<<B_, 1024, 0, stream>>>(keys, reg, props, candBox, obox, score, label, rowIdx, cvFlag);
  k3_nms<<<B_, 1024, 0, stream>>>(obox, candBox, score, label, rowIdx, cvFlag, quant,
                                  o_db, o_ds, o_dl, o_dq, o_dv, detRow);
  k4_feat<<<B_ * DET_, 256, 0, stream>>>(feats, detRow, o_df);
}